// PGcn_14654428414616
// MI455X (gfx1250) — compile-verified
//
#include <hip/hip_runtime.h>
#include <math.h>

#define NUM_USERS 100000
#define NUM_ITEMS 50000
#define N_NODES   150000
#define DIM       64
#define NLAYERS   3
#define NNZ       2400000
#define BATCH     8192
#define NEG_SLOPE 0.1f
#define EPS_N     1e-12f

typedef __attribute__((ext_vector_type(2))) float v2f;
typedef __attribute__((ext_vector_type(8))) float v8f;

__device__ __forceinline__ float lrelu(float x) { return x >= 0.f ? x : NEG_SLOPE * x; }

// One wave (32 lanes) per node; lane handles dims [2*lane, 2*lane+1].
__global__ void k_init_norm(const float* __restrict__ upref,
                            const float* __restrict__ ipref,
                            float* __restrict__ p_cur,
                            float* __restrict__ acc)
{
    int wave = (blockIdx.x * blockDim.x + threadIdx.x) >> 5;
    int lane = threadIdx.x & 31;
    if (wave >= N_NODES) return;
    const float* src = (wave < NUM_USERS) ? (upref + (size_t)wave * DIM)
                                          : (ipref + (size_t)(wave - NUM_USERS) * DIM);
    float2 x = reinterpret_cast<const float2*>(src)[lane];
    float x0 = lrelu(x.x), x1 = lrelu(x.y);
    float ss = x0 * x0 + x1 * x1;
    #pragma unroll
    for (int off = 16; off >= 1; off >>= 1) ss += __shfl_xor(ss, off, 32);
    float inv = 1.f / fmaxf(sqrtf(ss), EPS_N);
    size_t base = (size_t)wave * DIM + 2 * lane;
    float y0 = x0 * inv, y1 = x1 * inv;
    p_cur[base] = y0; p_cur[base + 1] = y1;
    acc[base]   = y0; acc[base + 1]   = y1;
}

__global__ void k_zero(float4* __restrict__ p, int n4)
{
    int i = blockIdx.x * blockDim.x + threadIdx.x;
    if (i < n4) p[i] = make_float4(0.f, 0.f, 0.f, 0.f);
}

// One wave per edge: 256B coalesced gather of the source row (L2-resident),
// 64 f32 atomic adds (global_atomic_add_f32) into the destination row.
__global__ void k_scatter(const int* __restrict__ rows,
                          const int* __restrict__ cols,
                          const float* __restrict__ vals,
                          const float* __restrict__ p_cur,
                          float* __restrict__ p_next)
{
    int wave = (blockIdx.x * blockDim.x + threadIdx.x) >> 5;
    int lane = threadIdx.x & 31;
    if (wave >= NNZ) return;
    int   r = rows[wave];
    int   c = cols[wave];
    float v = vals[wave];
    float2 x = reinterpret_cast<const float2*>(p_cur + (size_t)c * DIM)[lane];
    float* dst = p_next + (size_t)r * DIM + 2 * lane;
    unsafeAtomicAdd(dst + 0, v * x.x);
    unsafeAtomicAdd(dst + 1, v * x.y);
}

__global__ void k_norm_acc(const float* __restrict__ p_next,
                           float* __restrict__ p_cur,
                           float* __restrict__ acc)
{
    int wave = (blockIdx.x * blockDim.x + threadIdx.x) >> 5;
    int lane = threadIdx.x & 31;
    if (wave >= N_NODES) return;
    size_t base = (size_t)wave * DIM + 2 * lane;
    float2 x = *reinterpret_cast<const float2*>(p_next + base);
    float x0 = lrelu(x.x), x1 = lrelu(x.y);
    float ss = x0 * x0 + x1 * x1;
    #pragma unroll
    for (int off = 16; off >= 1; off >>= 1) ss += __shfl_xor(ss, off, 32);
    float inv = 1.f / fmaxf(sqrtf(ss), EPS_N);
    float y0 = x0 * inv, y1 = x1 * inv;
    p_cur[base] = y0; p_cur[base + 1] = y1;
    acc[base] += y0;  acc[base + 1] += y1;
}

// Scoring via V_WMMA_F32_16X16X4_F32: one wave computes 16 scores as the
// diagonal of U_tile(16x64) x I_tile^T(64x16), K consumed 4 at a time.
// A frag (16x4): lane l holds row (l&15), K = 4k + 2*(l>>4) .. +1.
// B frag (4x16): lane l holds col (l&15), same K pair (layout mirrors A).
// D diag: i<8 -> lane i comp i ; i>=8 -> lane i+16 comp i-8.
__global__ void __launch_bounds__(64)
k_score(const float* __restrict__ acc,
        const int* __restrict__ users,
        const int* __restrict__ adj,
        const int* __restrict__ weak,
        const int* __restrict__ strong,
        float* __restrict__ out)
{
    int lane = threadIdx.x & 31;
    int tile = blockIdx.x * (blockDim.x >> 5) + (threadIdx.x >> 5); // 0..511
    int set  = blockIdx.y;                                         // 0..2
    const int* items = (set == 0) ? adj : (set == 1) ? weak : strong;

    int m    = lane & 15;
    int half = lane >> 4;
    int i    = tile * 16 + m; // pair index 0..8191
    const float* urow = acc + (size_t)users[i] * DIM;
    const float* irow = acc + (size_t)(NUM_USERS + items[i]) * DIM;

    v8f c = {0.f, 0.f, 0.f, 0.f, 0.f, 0.f, 0.f, 0.f};
    #pragma unroll
    for (int k = 0; k < 16; ++k) {
        int off = 4 * k + 2 * half;
        v2f a, b;
        a.x = urow[off]; a.y = urow[off + 1];
        b.x = irow[off]; b.y = irow[off + 1];
        c = __builtin_amdgcn_wmma_f32_16x16x4_f32(
                /*neg_a=*/false, a, /*neg_b=*/false, b,
                /*c_mod=*/(short)0, c, /*reuse_a=*/false, /*reuse_b=*/false);
    }

    float d = 0.f;
    int outIdx = -1;
    if (lane < 8)        { d = c[lane];      outIdx = tile * 16 + lane; }
    else if (lane >= 24) { d = c[lane - 24]; outIdx = tile * 16 + (lane - 16); }
    if (outIdx >= 0) {
        // merged = acc/4 for both sides -> dot/16
        float s = 1.f / (1.f + __expf(-d * (1.f / 16.f)));
        out[(size_t)set * BATCH + outIdx] = s;
    }
}

extern "C" void kernel_launch(void* const* d_in, const int* in_sizes, int n_in,
                              void* d_out, int out_size, void* d_ws, size_t ws_size,
                              hipStream_t stream)
{
    const int*   users  = (const int*)  d_in[0];
    const int*   adj    = (const int*)  d_in[1];
    const int*   weak   = (const int*)  d_in[2];
    const int*   strong = (const int*)  d_in[3];
    const int*   erow   = (const int*)  d_in[4];
    const int*   ecol   = (const int*)  d_in[5];
    const float* eval   = (const float*)d_in[6];
    const float* upref  = (const float*)d_in[7];
    const float* ipref  = (const float*)d_in[8];
    float* out = (float*)d_out;

    const size_t mat = (size_t)N_NODES * DIM;
    float* acc    = (float*)d_ws;
    float* p_cur  = acc   + mat;
    float* p_next = p_cur + mat;

    k_init_norm<<<N_NODES / 8, 256, 0, stream>>>(upref, ipref, p_cur, acc);

    const int n4 = (int)(mat / 4);
    for (int l = 0; l < NLAYERS; ++l) {
        k_zero<<<(n4 + 255) / 256, 256, 0, stream>>>((float4*)p_next, n4);
        k_scatter<<<NNZ / 8, 256, 0, stream>>>(erow, ecol, eval, p_cur, p_next);
        k_norm_acc<<<N_NODES / 8, 256, 0, stream>>>(p_next, p_cur, acc);
    }

    dim3 g(BATCH / 16 / 2, 3); // 256 blocks x 3 sets, 2 waves/block, 1 tile/wave
    k_score<<<g, 64, 0, stream>>>(acc, users, adj, weak, strong, out);
}